// Net3_29755533427161
// MI455X (gfx1250) — compile-verified
//
#include <hip/hip_runtime.h>

typedef __attribute__((ext_vector_type(2))) float v2f;
typedef __attribute__((ext_vector_type(8))) float v8f;
typedef __attribute__((ext_vector_type(4))) int   v4i;

#define N_NODES 100000
#define N_EDGES 3200000

// ---------------------------------------------------------------------------
// Kernel 1: aggr := x   (folds the (1+eps)*x term, eps=0, into the aggregate
// buffer so the final node GEMM is a pure  aggr @ nn_w^T + nn_b)
// ---------------------------------------------------------------------------
__global__ void gine_init(const float* __restrict__ x, float* __restrict__ aggr) {
    int i = blockIdx.x * blockDim.x + threadIdx.x;   // float4 granularity
    if (i < N_NODES * 4) {
        ((float4*)aggr)[i] = ((const float4*)x)[i];
    }
}

// ---------------------------------------------------------------------------
// Kernel 2: per-edge  emb = edge_attr @ lin1_w^T + lin1_b  via WMMA f32 16x16x4,
// then msg = relu(x[src] + emb) scattered with fp32 L2 atomics into aggr.
// One wave handles one 16-edge tile per iteration (E % 16 == 0, EXEC all-1s).
// Streaming operands (edge_attr, edge_index) are loaded non-temporally so the
// L2 keeps the x/aggr working set resident.
// ---------------------------------------------------------------------------
__global__ void __launch_bounds__(256)
gine_edge(const float* __restrict__ edge_attr,   // [E,8]
          const int*   __restrict__ ei,          // [2,E]
          const float* __restrict__ x,           // [N,16]
          const float* __restrict__ lin1_w,      // [16,8]
          const float* __restrict__ lin1_b,      // [16]
          float*       __restrict__ aggr)        // [N,16]
{
    const int lane   = threadIdx.x & 31;
    const int wave   = (blockIdx.x * blockDim.x + threadIdx.x) >> 5;
    const int nwaves = (gridDim.x * blockDim.x) >> 5;
    const int half   = lane >> 4;      // which K pair / which row half of C
    const int n      = lane & 15;      // output feature column / A row in tile
    const int kA     = half * 2;

    // B = lin1_w^T chunks (4x16 each), loop-invariant.
    const v2f b_lo = *(const v2f*)(lin1_w + n * 8 + kA);       // K = 0..3 chunk
    const v2f b_hi = *(const v2f*)(lin1_w + n * 8 + 4 + kA);   // K = 4..7 chunk
    const float bias = lin1_b[n];

    const int* __restrict__ srcp = ei;             // row 0: x_j source
    const int* __restrict__ dstp = ei + N_EDGES;   // row 1: aggregation target

    const int ntiles = N_EDGES / 16;               // 200000 exactly
    for (int t = wave; t < ntiles; t += nwaves) {
        const int e0 = t * 16;

        // ---- Phase 0: issue all streaming loads (non-temporal) -------------
        // A: 16 edges x 8 attrs. Lane holds row M = n, K pair {kA,kA+1}(+4).
        const size_t arow = (size_t)(e0 + n) * 8;
        const v2f a_lo = __builtin_nontemporal_load((const v2f*)(edge_attr + arow + kA));
        const v2f a_hi = __builtin_nontemporal_load((const v2f*)(edge_attr + arow + 4 + kA));

        // Indices for the 8 C rows this lane owns: rows half*8 .. half*8+7.
        const int ibase = e0 + half * 8;
        const v4i s0 = __builtin_nontemporal_load((const v4i*)(srcp + ibase));
        const v4i s1 = __builtin_nontemporal_load((const v4i*)(srcp + ibase + 4));
        const v4i d0 = __builtin_nontemporal_load((const v4i*)(dstp + ibase));
        const v4i d1 = __builtin_nontemporal_load((const v4i*)(dstp + ibase + 4));

        // ---- Phase 1: WMMA edge embedding (bias pre-broadcast in C) --------
        v8f c;
        #pragma unroll
        for (int j = 0; j < 8; ++j) c[j] = bias;

        c = __builtin_amdgcn_wmma_f32_16x16x4_f32(false, a_lo, false, b_lo,
                                                  (short)0, c, false, false);
        c = __builtin_amdgcn_wmma_f32_16x16x4_f32(false, a_hi, false, b_hi,
                                                  (short)0, c, false, false);

        // ---- Phase 2: issue all 8 x-gathers before consuming any -----------
        int sd[8], dd[8];
        #pragma unroll
        for (int j = 0; j < 8; ++j) {
            sd[j] = (j < 4) ? s0[j] : s1[j - 4];
            dd[j] = (j < 4) ? d0[j] : d1[j - 4];
        }
        float xv[8];
        #pragma unroll
        for (int j = 0; j < 8; ++j) {
            xv[j] = x[sd[j] * 16 + n];          // L2-resident gather
        }

        // ---- Phase 3: ReLU + scatter-add (hardware fp32 atomics in L2) -----
        #pragma unroll
        for (int j = 0; j < 8; ++j) {
            float msg = xv[j] + c[j];
            msg = msg > 0.0f ? msg : 0.0f;
            unsafeAtomicAdd(&aggr[dd[j] * 16 + n], msg);
        }
    }
}

// ---------------------------------------------------------------------------
// Kernel 3: out = aggr @ nn_w^T + nn_b   ([N,16] x [16,32] -> [N,32])
// Per 16-node tile: 4 K-chunks x 2 column tiles = 8 WMMAs.
// Output is written once and never re-read -> non-temporal stores.
// ---------------------------------------------------------------------------
__global__ void __launch_bounds__(256)
gine_node(const float* __restrict__ aggr,   // [N,16] = x + sum(msg)
          const float* __restrict__ nn_w,   // [32,16]
          const float* __restrict__ nn_b,   // [32]
          float*       __restrict__ out)    // [N,32]
{
    const int lane   = threadIdx.x & 31;
    const int wave   = (blockIdx.x * blockDim.x + threadIdx.x) >> 5;
    const int nwaves = (gridDim.x * blockDim.x) >> 5;
    const int half   = lane >> 4;
    const int n      = lane & 15;

    // B chunks for output columns n (tile 0) and n+16 (tile 1); invariant.
    v2f b0[4], b1[4];
    #pragma unroll
    for (int kk = 0; kk < 4; ++kk) {
        const int k = kk * 4 + half * 2;
        b0[kk] = *(const v2f*)(nn_w + n * 16 + k);
        b1[kk] = *(const v2f*)(nn_w + (n + 16) * 16 + k);
    }
    const float bias0 = nn_b[n];
    const float bias1 = nn_b[n + 16];

    const int ntiles = N_NODES / 16;    // 6250 exactly
    for (int t = wave; t < ntiles; t += nwaves) {
        const int node0 = t * 16;
        v2f a[4];
        #pragma unroll
        for (int kk = 0; kk < 4; ++kk) {
            const int k = kk * 4 + half * 2;
            a[kk] = *(const v2f*)(aggr + (size_t)(node0 + n) * 16 + k);
        }

        v8f c0, c1;
        #pragma unroll
        for (int j = 0; j < 8; ++j) { c0[j] = bias0; c1[j] = bias1; }

        #pragma unroll
        for (int kk = 0; kk < 4; ++kk) {
            c0 = __builtin_amdgcn_wmma_f32_16x16x4_f32(false, a[kk], false, b0[kk],
                                                       (short)0, c0, false, false);
            c1 = __builtin_amdgcn_wmma_f32_16x16x4_f32(false, a[kk], false, b1[kk],
                                                       (short)0, c1, false, false);
        }

        #pragma unroll
        for (int j = 0; j < 8; ++j) {
            const int row = node0 + j + half * 8;
            __builtin_nontemporal_store(c0[j], &out[row * 32 + n]);
            __builtin_nontemporal_store(c1[j], &out[row * 32 + n + 16]);
        }
    }
}

// ---------------------------------------------------------------------------
extern "C" void kernel_launch(void* const* d_in, const int* in_sizes, int n_in,
                              void* d_out, int out_size, void* d_ws, size_t ws_size,
                              hipStream_t stream) {
    const float* x         = (const float*)d_in[0];   // [N,16]
    const int*   ei        = (const int*)  d_in[1];   // [2,E]
    const float* edge_attr = (const float*)d_in[2];   // [E,8]
    const float* lin1_w    = (const float*)d_in[3];   // [16,8]
    const float* lin1_b    = (const float*)d_in[4];   // [16]
    const float* nn_w      = (const float*)d_in[5];   // [32,16]
    const float* nn_b      = (const float*)d_in[6];   // [32]
    float* out  = (float*)d_out;
    float* aggr = (float*)d_ws;                       // N*16 floats = 6.4 MB

    (void)in_sizes; (void)n_in; (void)out_size; (void)ws_size;

    gine_init<<<(N_NODES * 4 + 255) / 256, 256, 0, stream>>>(x, aggr);
    gine_edge<<<2048, 256, 0, stream>>>(edge_attr, ei, x, lin1_w, lin1_b, aggr);
    gine_node<<<512, 256, 0, stream>>>(aggr, nn_w, nn_b, out);
}